// ReUpSampler_14886356648192
// MI455X (gfx1250) — compile-verified
//
#include <hip/hip_runtime.h>
#include <hip/hip_bf16.h>

typedef __attribute__((ext_vector_type(2))) float v2f;
typedef __attribute__((ext_vector_type(8))) float v8f;

// ---------------------------------------------------------------------------
// Kernel 1: group-norm statistics. One block per (batch, group).
// stats[bg*2] = mean, stats[bg*2+1] = rsqrt(var + eps)
// ---------------------------------------------------------------------------
__global__ __launch_bounds__(256)
void gn_stats_kernel(const float* __restrict__ x, float* __restrict__ stats,
                     int C, int HW, int ng, float eps) {
  int bg = blockIdx.x;
  int cg = C / ng;
  int b = bg / ng, g = bg % ng;
  const float* base = x + ((size_t)b * C + (size_t)g * cg) * HW;
  long n = (long)cg * HW;
  float s = 0.f, ss = 0.f;
  for (long i = threadIdx.x; i < n; i += blockDim.x) {
    float v = base[i];
    s += v; ss += v * v;
  }
  __shared__ float sh_s[256], sh_ss[256];
  sh_s[threadIdx.x] = s; sh_ss[threadIdx.x] = ss;
  __syncthreads();
  for (int off = 128; off > 0; off >>= 1) {
    if ((int)threadIdx.x < off) {
      sh_s[threadIdx.x]  += sh_s[threadIdx.x + off];
      sh_ss[threadIdx.x] += sh_ss[threadIdx.x + off];
    }
    __syncthreads();
  }
  if (threadIdx.x == 0) {
    float inv_n = 1.0f / (float)n;
    float mean = sh_s[0] * inv_n;
    float var  = sh_ss[0] * inv_n - mean * mean;
    stats[bg * 2]     = mean;
    stats[bg * 2 + 1] = rsqrtf(var + eps);
  }
}

// ---------------------------------------------------------------------------
// Kernel 2: hf path. Normalize hf on the fly, compute 8-neighbor cosine
// similarity + conv1x1(sim, 8->8) and conv1x1(hfn, 32->8); store both in the
// pixel-unshuffled (B,32,128,128) layout so downstream indexing is direct.
// 16x16 pixel tile per block, channels in chunks of 16 through LDS.
// ---------------------------------------------------------------------------
__global__ __launch_bounds__(256)
void hf_feat_kernel(const float* __restrict__ hf, const float* __restrict__ stats,
                    const float* __restrict__ gamma, const float* __restrict__ beta,
                    const float* __restrict__ wc, const float* __restrict__ bc,
                    const float* __restrict__ wds, const float* __restrict__ bds,
                    float* __restrict__ raw_u, float* __restrict__ gate_u) {
  const int C = 32, H = 256, W = 256, CHUNK = 16;
  __shared__ float tile[CHUNK][18][20];
  __shared__ float s_wc[64], s_bc[8], s_wds[256], s_bds[8], s_sc[32], s_sh[32];
  int tid = threadIdx.x;
  int b = blockIdx.z;
  int tx = tid & 15, ty = tid >> 4;
  int x0 = blockIdx.x * 16, y0 = blockIdx.y * 16;

  s_wds[tid & 255] = wds[tid & 255];
  if (tid < 64) s_wc[tid] = wc[tid];
  if (tid >= 64 && tid < 72)  s_bc[tid - 64]  = bc[tid - 64];
  if (tid >= 72 && tid < 80)  s_bds[tid - 72] = bds[tid - 72];
  if (tid >= 96 && tid < 128) {
    int c = tid - 96;
    int g = c >> 3;
    float mean = stats[(b * 4 + g) * 2];
    float rstd = stats[(b * 4 + g) * 2 + 1];
    float sc = rstd * gamma[c];
    s_sc[c] = sc;
    s_sh[c] = beta[c] - mean * sc;
  }

  float dot[9], nrm[9], t2[8];
#pragma unroll
  for (int j = 0; j < 9; ++j) { dot[j] = 0.f; nrm[j] = 0.f; }
  const float* inb = hf + (size_t)b * C * H * W;
  __syncthreads();
#pragma unroll
  for (int o = 0; o < 8; ++o) t2[o] = s_bds[o];

  for (int ch = 0; ch < C; ch += CHUNK) {
    for (int idx = tid; idx < CHUNK * 18 * 18; idx += 256) {
      int cl = idx / 324, rem = idx % 324;
      int ly = rem / 18, lx = rem % 18;
      int gy = y0 + ly - 1, gx = x0 + lx - 1;
      int c = ch + cl;
      float v = 0.f;
      if (gy >= 0 && gy < H && gx >= 0 && gx < W)
        v = inb[(size_t)c * H * W + (size_t)gy * W + gx] * s_sc[c] + s_sh[c];
      tile[cl][ly][lx] = v;
    }
    __syncthreads();
#pragma unroll
    for (int cl = 0; cl < CHUNK; ++cl) {
      float ctr = tile[cl][ty + 1][tx + 1];
#pragma unroll
      for (int j = 0; j < 9; ++j) {
        float nb = tile[cl][ty + j / 3][tx + j % 3];
        dot[j] += ctr * nb;
        nrm[j] += nb * nb;
      }
      int c = ch + cl;
#pragma unroll
      for (int o = 0; o < 8; ++o) t2[o] += s_wds[o * 32 + c] * ctr;
    }
    __syncthreads();
  }

  float cn = sqrtf(nrm[4]);
  float sim8[8];
  int jj = 0;
#pragma unroll
  for (int j = 0; j < 9; ++j) {
    if (j == 4) continue;
    sim8[jj++] = dot[j] / fmaxf(cn * sqrtf(nrm[j]), 1e-8f);
  }
  int gx = x0 + tx, gy = y0 + ty;
  int xl = gx >> 1, yl = gy >> 1;
  int sidx = ((gy & 1) << 1) | (gx & 1);
  size_t ob = (size_t)b * 32 * (128 * 128) + (size_t)yl * 128 + xl;
#pragma unroll
  for (int o = 0; o < 8; ++o) {
    float t1 = s_bc[o];
#pragma unroll
    for (int j = 0; j < 8; ++j) t1 += s_wc[o * 8 + j] * sim8[j];
    size_t coff = (size_t)(o * 4 + sidx) * (128 * 128);
    raw_u[ob + coff]  = t1;
    gate_u[ob + coff] = t2[o];
  }
}

// ---------------------------------------------------------------------------
// Kernel 3: lf 8-neighbor cosine similarity on normalized lf, stored
// channel-planar (B,8,128,128) so it can feed the WMMA B-matrix loads.
// ---------------------------------------------------------------------------
__global__ __launch_bounds__(256)
void lf_sim_kernel(const float* __restrict__ lf, const float* __restrict__ stats,
                   const float* __restrict__ gamma, const float* __restrict__ beta,
                   float* __restrict__ sim_out) {
  const int C = 64, H = 128, W = 128, CHUNK = 16;
  __shared__ float tile[CHUNK][18][20];
  __shared__ float s_sc[64], s_sh[64];
  int tid = threadIdx.x;
  int b = blockIdx.z;
  int tx = tid & 15, ty = tid >> 4;
  int x0 = blockIdx.x * 16, y0 = blockIdx.y * 16;
  if (tid < 64) {
    int g = tid >> 3;
    float mean = stats[(b * 8 + g) * 2];
    float rstd = stats[(b * 8 + g) * 2 + 1];
    float sc = rstd * gamma[tid];
    s_sc[tid] = sc;
    s_sh[tid] = beta[tid] - mean * sc;
  }
  __syncthreads();
  float dot[9], nrm[9];
#pragma unroll
  for (int j = 0; j < 9; ++j) { dot[j] = 0.f; nrm[j] = 0.f; }
  const float* inb = lf + (size_t)b * C * H * W;
  for (int ch = 0; ch < C; ch += CHUNK) {
    for (int idx = tid; idx < CHUNK * 18 * 18; idx += 256) {
      int cl = idx / 324, rem = idx % 324;
      int ly = rem / 18, lx = rem % 18;
      int gy = y0 + ly - 1, gx = x0 + lx - 1;
      int c = ch + cl;
      float v = 0.f;
      if (gy >= 0 && gy < H && gx >= 0 && gx < W)
        v = inb[(size_t)c * H * W + (size_t)gy * W + gx] * s_sc[c] + s_sh[c];
      tile[cl][ly][lx] = v;
    }
    __syncthreads();
#pragma unroll
    for (int cl = 0; cl < CHUNK; ++cl) {
      float ctr = tile[cl][ty + 1][tx + 1];
#pragma unroll
      for (int j = 0; j < 9; ++j) {
        float nb = tile[cl][ty + j / 3][tx + j % 3];
        dot[j] += ctr * nb;
        nrm[j] += nb * nb;
      }
    }
    __syncthreads();
  }
  float cn = sqrtf(nrm[4]);
  int gx = x0 + tx, gy = y0 + ty;
  size_t ob = (size_t)b * 8 * (H * W) + (size_t)gy * W + gx;
  int jj = 0;
#pragma unroll
  for (int j = 0; j < 9; ++j) {
    if (j == 4) continue;
    sim_out[ob + (size_t)jj * (H * W)] = dot[j] / fmaxf(cn * sqrtf(nrm[j]), 1e-8f);
    jj++;
  }
}

// ---------------------------------------------------------------------------
// Kernel 4: offset field via f32 WMMA. Each wave owns 16 pixels.
//   gate_pre = ds_w(32x64) @ lfn        -> 2 M-tiles x 16 K-steps  (32 WMMA)
//   raw_lf   = lf_conv_w(32x8) @ lf_sim -> 2 M-tiles x 2 K-steps   (4 WMMA)
// A layout (16x4 f32): lanes 0-15 hold K={k0,k0+1}, lanes 16-31 K={k0+2,k0+3}.
// B layout (4x16) / D (16x16): row-striped across lanes, half-wave +offset.
// ---------------------------------------------------------------------------
__global__ __launch_bounds__(256)
void offset_wmma_kernel(const float* __restrict__ lf,
                        const float* __restrict__ lf_stats,
                        const float* __restrict__ gn_g, const float* __restrict__ gn_b,
                        const float* __restrict__ ds_w, const float* __restrict__ ds_b,
                        const float* __restrict__ lf_conv_w, const float* __restrict__ lf_conv_b,
                        const float* __restrict__ lf_sim,
                        const float* __restrict__ raw_hf_u, const float* __restrict__ gate_hf_u,
                        float* __restrict__ offset_out) {
  const int HW = 128 * 128;
  __shared__ float s_dsw[32 * 64];
  __shared__ float s_lcw[32 * 8];
  __shared__ float s_dsb[32], s_lcb[32], s_sc[64], s_sh[64];
  int tid = threadIdx.x;
  int b = blockIdx.x >> 7;            // 128 blocks per batch
  int blk_in_b = blockIdx.x & 127;

  for (int i = tid; i < 2048; i += 256) s_dsw[i] = ds_w[i];
  s_lcw[tid & 255] = lf_conv_w[tid & 255];
  if (tid < 32) { s_dsb[tid] = ds_b[tid]; s_lcb[tid] = lf_conv_b[tid]; }
  if (tid >= 64 && tid < 128) {
    int c = tid - 64;
    int g = c >> 3;
    float mean = lf_stats[(b * 8 + g) * 2];
    float rstd = lf_stats[(b * 8 + g) * 2 + 1];
    float sc = rstd * gn_g[c];
    s_sc[c] = sc;
    s_sh[c] = gn_b[c] - mean * sc;
  }
  __syncthreads();

  int wave = tid >> 5, lane = tid & 31;
  int half = lane >> 4, ln = lane & 15;
  int pix0 = (blk_in_b * 8 + wave) * 16;
  const float* lfb  = lf     + (size_t)b * 64 * HW;
  const float* simb = lf_sim + (size_t)b * 8 * HW;

  v8f acc0, acc1, r0, r1;
#pragma unroll
  for (int v = 0; v < 8; ++v) {
    int m = v + 8 * half;               // D row for this lane-half
    acc0[v] = s_dsb[m];      acc1[v] = s_dsb[m + 16];
    r0[v]   = s_lcb[m];      r1[v]   = s_lcb[m + 16];
  }

  // GEMM 1: gate_pre = ds_w @ lfn, K = 64
#pragma unroll
  for (int k0 = 0; k0 < 64; k0 += 4) {
    int kb = k0 + half * 2;
    v2f bmat, a0, a1;
    float x0v = lfb[(size_t)kb * HW + pix0 + ln];
    float x1v = lfb[(size_t)(kb + 1) * HW + pix0 + ln];
    bmat[0] = x0v * s_sc[kb]     + s_sh[kb];
    bmat[1] = x1v * s_sc[kb + 1] + s_sh[kb + 1];
    a0[0] = s_dsw[ln * 64 + kb];        a0[1] = s_dsw[ln * 64 + kb + 1];
    a1[0] = s_dsw[(ln + 16) * 64 + kb]; a1[1] = s_dsw[(ln + 16) * 64 + kb + 1];
    acc0 = __builtin_amdgcn_wmma_f32_16x16x4_f32(false, a0, false, bmat, (short)0, acc0, false, false);
    acc1 = __builtin_amdgcn_wmma_f32_16x16x4_f32(false, a1, false, bmat, (short)0, acc1, false, false);
  }

  // GEMM 2: raw_lf = lf_conv_w @ lf_sim, K = 8
#pragma unroll
  for (int k0 = 0; k0 < 8; k0 += 4) {
    int kb = k0 + half * 2;
    v2f bmat, a0, a1;
    bmat[0] = simb[(size_t)kb * HW + pix0 + ln];
    bmat[1] = simb[(size_t)(kb + 1) * HW + pix0 + ln];
    a0[0] = s_lcw[ln * 8 + kb];        a0[1] = s_lcw[ln * 8 + kb + 1];
    a1[0] = s_lcw[(ln + 16) * 8 + kb]; a1[1] = s_lcw[(ln + 16) * 8 + kb + 1];
    r0 = __builtin_amdgcn_wmma_f32_16x16x4_f32(false, a0, false, bmat, (short)0, r0, false, false);
    r1 = __builtin_amdgcn_wmma_f32_16x16x4_f32(false, a1, false, bmat, (short)0, r1, false, false);
  }

  // Epilogue: + hf contributions, sigmoid gate, init_pos, store offset.
  size_t base = (size_t)b * 32 * HW;
  int n = pix0 + ln;
#pragma unroll
  for (int v = 0; v < 8; ++v) {
    int m = v + 8 * half;
#pragma unroll
    for (int t = 0; t < 2; ++t) {
      int c = m + 16 * t;
      size_t idx = base + (size_t)c * HW + n;
      float gpre = (t ? acc1[v] : acc0[v]) + gate_hf_u[idx];
      float rv   = (t ? r1[v]   : r0[v])   + raw_hf_u[idx];
      float gate = 1.f / (1.f + __expf(-gpre));
      int p = c & 3;
      float init = (c < 16) ? ((p & 1)  ? 0.25f : -0.25f)
                            : ((p >> 1) ? 0.25f : -0.25f);
      offset_out[idx] = rv * gate + init;
    }
  }
}

// ---------------------------------------------------------------------------
// Kernel 5: grouped bilinear sampling. grid_sample(align_corners=False,
// border) with the reference's coord normalization simplifies exactly to
// sample position (xl + offx, yl + offy) clamped to [0, 127].
// One thread per (b, group, out-pixel); 16 channels reuse the weights.
// ---------------------------------------------------------------------------
__global__ __launch_bounds__(256)
void sample_kernel(const float* __restrict__ lf, const float* __restrict__ offset,
                   float* __restrict__ out) {
  const int W = 128, HW = 128 * 128, sW = 256, sHW = 256 * 256;
  int flat = blockIdx.x * 256 + threadIdx.x;   // 4*4*256*256 = 1,048,576
  int xo = flat & 255;
  int yo = (flat >> 8) & 255;
  int g  = (flat >> 16) & 3;
  int b  = flat >> 18;
  int xl = xo >> 1, yl = yo >> 1;
  int p = ((yo & 1) << 1) | (xo & 1);
  size_t obase = (size_t)b * 32 * HW + (size_t)yl * W + xl;
  float offx = offset[obase + (size_t)(g * 4 + p) * HW];
  float offy = offset[obase + (size_t)(16 + g * 4 + p) * HW];
  float sx = fminf(fmaxf((float)xl + offx, 0.f), 127.f);
  float sy = fminf(fmaxf((float)yl + offy, 0.f), 127.f);
  float x0f = floorf(sx), y0f = floorf(sy);
  float wx = sx - x0f, wy = sy - y0f;
  int x0 = (int)x0f, y0 = (int)y0f;
  int x1 = min(x0 + 1, 127), y1 = min(y0 + 1, 127);
  float w00 = (1.f - wx) * (1.f - wy), w01 = wx * (1.f - wy);
  float w10 = (1.f - wx) * wy,         w11 = wx * wy;
  int i00 = y0 * W + x0, i01 = y0 * W + x1, i10 = y1 * W + x0, i11 = y1 * W + x1;
  const float* src = lf + ((size_t)b * 64 + g * 16) * HW;
  float* dst = out + ((size_t)b * 64 + g * 16) * sHW + (size_t)yo * sW + xo;
#pragma unroll
  for (int cl = 0; cl < 16; ++cl) {
    const float* s = src + (size_t)cl * HW;
    dst[(size_t)cl * sHW] = s[i00] * w00 + s[i01] * w01 + s[i10] * w10 + s[i11] * w11;
  }
}

// ---------------------------------------------------------------------------
extern "C" void kernel_launch(void* const* d_in, const int* in_sizes, int n_in,
                              void* d_out, int out_size, void* d_ws, size_t ws_size,
                              hipStream_t stream) {
  const float* hf        = (const float*)d_in[0];
  const float* lf        = (const float*)d_in[1];
  const float* lf_conv_w = (const float*)d_in[2];
  const float* lf_conv_b = (const float*)d_in[3];
  const float* hf_conv_w = (const float*)d_in[4];
  const float* hf_conv_b = (const float*)d_in[5];
  const float* ds_w      = (const float*)d_in[6];
  const float* ds_b      = (const float*)d_in[7];
  const float* hds_w     = (const float*)d_in[8];
  const float* hds_b     = (const float*)d_in[9];
  const float* gn_hr_g   = (const float*)d_in[10];
  const float* gn_hr_b   = (const float*)d_in[11];
  const float* gn_lr_g   = (const float*)d_in[12];
  const float* gn_lr_b   = (const float*)d_in[13];
  float* out = (float*)d_out;

  float* ws        = (float*)d_ws;
  float* hf_stats  = ws;                       // 32 floats
  float* lf_stats  = ws + 32;                  // 64 floats
  float* raw_hf_u  = ws + 96;                  // 4*32*128*128
  float* gate_hf_u = raw_hf_u + 2097152;       // 4*32*128*128
  float* lf_simbuf = gate_hf_u + 2097152;      // 4*8*128*128
  float* offsetbuf = lf_simbuf + 524288;       // 4*32*128*128

  gn_stats_kernel<<<16, 256, 0, stream>>>(hf, hf_stats, 32, 256 * 256, 4, 1e-5f);
  gn_stats_kernel<<<32, 256, 0, stream>>>(lf, lf_stats, 64, 128 * 128, 8, 1e-5f);

  dim3 gh(16, 16, 4);
  hf_feat_kernel<<<gh, 256, 0, stream>>>(hf, hf_stats, gn_hr_g, gn_hr_b,
                                         hf_conv_w, hf_conv_b, hds_w, hds_b,
                                         raw_hf_u, gate_hf_u);
  dim3 gl(8, 8, 4);
  lf_sim_kernel<<<gl, 256, 0, stream>>>(lf, lf_stats, gn_lr_g, gn_lr_b, lf_simbuf);

  offset_wmma_kernel<<<512, 256, 0, stream>>>(lf, lf_stats, gn_lr_g, gn_lr_b,
                                              ds_w, ds_b, lf_conv_w, lf_conv_b,
                                              lf_simbuf, raw_hf_u, gate_hf_u, offsetbuf);

  sample_kernel<<<4096, 256, 0, stream>>>(lf, offsetbuf, out);
}